// DiscreteGraphLearning_77653008711867
// MI455X (gfx1250) — compile-verified
//
#include <hip/hip_runtime.h>
#include <hip/hip_bf16.h>
#include <math.h>

// ---------------- problem constants ----------------
#define Nn   307
#define Tt   6000
#define C1_  8
#define C2_  16
#define L1_  5991          // Tt - 9
#define L2_  5982          // L1_ - 9
#define ED_  96
#define MP_  320           // padded M (20 tiles of 16)
#define KCH  96            // GEMM K-chunk staged through LDS
#define EPSB 1e-5f

typedef __attribute__((ext_vector_type(2))) float v2f;
typedef __attribute__((ext_vector_type(8))) float v8f;

// ---------------- 1) transpose node_feats [T,N] -> xT [N,T] ----------------
__global__ void k_transpose(const float* __restrict__ nf, float* __restrict__ xT) {
    __shared__ float tile[32][33];
    int t0 = blockIdx.x * 32, n0 = blockIdx.y * 32;
    int tx = threadIdx.x, ty = threadIdx.y;           // 32 x 8
    for (int r = ty; r < 32; r += 8) {
        int t = t0 + r, n = n0 + tx;
        tile[r][tx] = (t < Tt && n < Nn) ? nf[(size_t)t * Nn + n] : 0.f;
    }
    __syncthreads();
    for (int r = ty; r < 32; r += 8) {
        int n = n0 + r, t = t0 + tx;
        if (n < Nn && t < Tt) xT[(size_t)n * Tt + t] = tile[tx][r];
    }
}

// ---------------- 2) conv1 (1->8, k=10) + ReLU -> y1 [N,8,L1] ----------------
__global__ void __launch_bounds__(256) k_conv1(const float* __restrict__ xT,
                                               const float* __restrict__ w,
                                               const float* __restrict__ b,
                                               float* __restrict__ y1) {
    __shared__ float lx[256 + 10];
    __shared__ float lw[C1_ * 10];
    __shared__ float lb[C1_];
    int n = blockIdx.y, l0 = blockIdx.x * 256, tid = threadIdx.x;
    if (tid < C1_ * 10) lw[tid] = w[tid];
    if (tid < C1_)      lb[tid] = b[tid];
    for (int i = tid; i < 256 + 10; i += 256) {
        int t = l0 + i;
        lx[i] = (t < Tt) ? xT[(size_t)n * Tt + t] : 0.f;
    }
    __syncthreads();
    int l = l0 + tid;
    if (l < L1_) {
        float win[10];
#pragma unroll
        for (int k = 0; k < 10; k++) win[k] = lx[tid + k];
#pragma unroll
        for (int c = 0; c < C1_; c++) {
            float acc = lb[c];
#pragma unroll
            for (int k = 0; k < 10; k++) acc = fmaf(lw[c * 10 + k], win[k], acc);
            y1[(size_t)n * C1_ * L1_ + (size_t)c * L1_ + l] = fmaxf(acc, 0.f);
        }
    }
}

// ------- BN train-mode stats (deterministic): one block per channel -------
// computes a[c] = g/sqrt(var+eps), d[c] = b - a*mean  (batch stats over N*L)
__global__ void __launch_bounds__(256) k_bnstats(const float* __restrict__ y, int C, int L,
                                                 const float* __restrict__ g,
                                                 const float* __restrict__ b,
                                                 float* __restrict__ a, float* __restrict__ d) {
    int c = blockIdx.x, tid = threadIdx.x;
    float s = 0.f, ss = 0.f;
    for (int n = 0; n < Nn; n++) {
        const float* row = y + (size_t)n * C * L + (size_t)c * L;
        for (int l = tid; l < L; l += 256) { float v = row[l]; s += v; ss += v * v; }
    }
    __shared__ float rs[256], rq[256];
    rs[tid] = s; rq[tid] = ss;
    __syncthreads();
    for (int o = 128; o > 0; o >>= 1) {
        if (tid < o) { rs[tid] += rs[tid + o]; rq[tid] += rq[tid + o]; }
        __syncthreads();
    }
    if (tid == 0) {
        float cnt = (float)Nn * (float)L;
        float m = rs[0] / cnt;
        float v = rq[0] / cnt - m * m;
        float ai = g[c] * rsqrtf(v + EPSB);
        a[c] = ai; d[c] = b[c] - ai * m;
    }
}

// ------- 3) conv2 (8->16, k=10) over BN1 output + ReLU -> y2 [N,16,L2] -------
// BN1 affine folded into weights/bias: w' = a1[c1]*w, b' = b2 + sum d1[c1]*w
__global__ void __launch_bounds__(256) k_conv2(const float* __restrict__ y1,
                                               const float* __restrict__ w2,
                                               const float* __restrict__ b2,
                                               const float* __restrict__ a1,
                                               const float* __restrict__ d1,
                                               float* __restrict__ y2) {
    __shared__ float ly[C1_ * 266];
    __shared__ float lw[C2_ * C1_ * 10];
    __shared__ float lb[C2_];
    int tid = threadIdx.x;
    for (int i = tid; i < C2_ * C1_ * 10; i += 256) {
        int c1 = (i / 10) % C1_;
        lw[i] = a1[c1] * w2[i];
    }
    if (tid < C2_) {
        float acc = b2[tid];
        for (int c1 = 0; c1 < C1_; c1++) {
            float dd = d1[c1];
            for (int k = 0; k < 10; k++) acc += dd * w2[(tid * C1_ + c1) * 10 + k];
        }
        lb[tid] = acc;
    }
    int n = blockIdx.y, l0 = blockIdx.x * 256;
    for (int i = tid; i < C1_ * 266; i += 256) {
        int c1 = i / 266, off = i - c1 * 266;
        int l = l0 + off;
        ly[i] = (l < L1_) ? y1[(size_t)n * C1_ * L1_ + (size_t)c1 * L1_ + l] : 0.f;
    }
    __syncthreads();
    int l = l0 + tid;
    if (l < L2_) {
        float acc[C2_];
#pragma unroll
        for (int c2 = 0; c2 < C2_; c2++) acc[c2] = lb[c2];
        for (int c1 = 0; c1 < C1_; c1++) {
            float win[10];
#pragma unroll
            for (int k = 0; k < 10; k++) win[k] = ly[c1 * 266 + tid + k];
#pragma unroll
            for (int c2 = 0; c2 < C2_; c2++) {
#pragma unroll
                for (int k = 0; k < 10; k++)
                    acc[c2] = fmaf(lw[(c2 * C1_ + c1) * 10 + k], win[k], acc[c2]);
            }
        }
        size_t base = (size_t)n * C2_ * L2_ + l;
#pragma unroll
        for (int c2 = 0; c2 < C2_; c2++) y2[base + (size_t)c2 * L2_] = fmaxf(acc[c2], 0.f);
    }
}

// ---- 4) wsum[c][e] = sum_l fc_w[c*L2+l][e]  (folds BN2 shift into FC bias) ----
__global__ void k_wsum(const float* __restrict__ fcw, float* __restrict__ wsum) {
    int c = blockIdx.x, e = threadIdx.x;     // 96 threads
    float s = 0.f;
    const float* p = fcw + (size_t)c * L2_ * ED_ + e;
    for (int l = 0; l < L2_; l++) s += p[(size_t)l * ED_];
    wsum[c * ED_ + e] = s;
}

__global__ void k_biasfc(const float* __restrict__ wsum, const float* __restrict__ d2,
                         const float* __restrict__ fcb, float* __restrict__ biasfc) {
    int e = threadIdx.x;
    float s = fcb[e];
    for (int c = 0; c < C2_; c++) s += d2[c] * wsum[c * ED_ + e];
    biasfc[e] = s;
}

// ---------------- 5) big FC GEMM with f32 WMMA (16x16x4) ----------------
// grid: (channel c, M-tile mt). Each block: 6 waves, wave w -> N-cols [16w,16w+16).
// Computes partial[c][m][e] = sum_{l in channel c} a2[c]*y2[m][c][l] * fc_w[c*L2+l][e]
__global__ void __launch_bounds__(192) k_gemm_wmma(const float* __restrict__ y2,
                                                   const float* __restrict__ fcw,
                                                   const float* __restrict__ a2,
                                                   float* __restrict__ part) {
    __shared__ float lA[16 * KCH];
    __shared__ float lB[KCH * ED_];
    int c = blockIdx.x, mt = blockIdx.y;
    int m0 = mt * 16;
    int tid = threadIdx.x;
    int wave = tid >> 5, lane = tid & 31;
    int lane16 = lane & 15, hv = lane >> 4;   // lane half selects K {0,1} vs {2,3}
    int ecol = wave * 16;
    float a2c = a2[c];
    v8f acc = {};
    const float* Abase = y2 + (size_t)c * L2_;
    const float* Bbase = fcw + (size_t)c * L2_ * ED_;

    for (int l0 = 0; l0 < L2_; l0 += KCH) {
        // stage A tile [16 rows x KCH] (BN2 scale applied, zero-padded)
        for (int i = tid; i < 16 * KCH; i += 192) {
            int r = i / KCH, col = i - r * KCH;
            int n = m0 + r, l = l0 + col;
            float v = 0.f;
            if (n < Nn && l < L2_) v = a2c * Abase[(size_t)n * C2_ * L2_ + l];
            lA[i] = v;
        }
        // stage B tile [KCH rows x 96 cols] (zero-padded tail)
        for (int i = tid; i < KCH * ED_; i += 192) {
            int r = i / ED_, e = i - r * ED_;
            int l = l0 + r;
            lB[i] = (l < L2_) ? Bbase[(size_t)l * ED_ + e] : 0.f;
        }
        __syncthreads();
#pragma unroll
        for (int k4 = 0; k4 < KCH / 4; k4++) {
            int kb = k4 * 4 + hv * 2;
            v2f a = *(const v2f*)&lA[lane16 * KCH + kb];      // A[M=lane16][K=kb,kb+1]
            v2f b;
            b.x = lB[kb * ED_ + ecol + lane16];               // B[K=kb  ][N=lane16]
            b.y = lB[(kb + 1) * ED_ + ecol + lane16];         // B[K=kb+1][N=lane16]
            acc = __builtin_amdgcn_wmma_f32_16x16x4_f32(
                false, a, false, b, (short)0, acc, false, false);
        }
        __syncthreads();
    }
    // C/D layout: VGPR r -> M = r + 8*half, N = lane16
    float* dst = part + (size_t)c * MP_ * ED_;
#pragma unroll
    for (int r = 0; r < 8; r++) {
        int m = m0 + r + hv * 8;
        dst[(size_t)m * ED_ + ecol + lane16] = acc[r];
    }
}

// ---------------- 6) combine partials + bias + ReLU -> x [N,96] ----------------
__global__ void k_combine(const float* __restrict__ part, const float* __restrict__ biasfc,
                          float* __restrict__ x) {
    int idx = blockIdx.x * 256 + threadIdx.x;
    if (idx >= Nn * ED_) return;
    int n = idx / ED_, e = idx - n * ED_;
    float s = biasfc[e];
    for (int c = 0; c < C2_; c++) s += part[(size_t)c * MP_ * ED_ + (size_t)n * ED_ + e];
    x[idx] = fmaxf(s, 0.f);
}

// ---------------- 7) BN3 (over nodes) -> affine A3,B3 ----------------
__global__ void k_bn3(const float* __restrict__ x, const float* __restrict__ g,
                      const float* __restrict__ b, float* __restrict__ A3,
                      float* __restrict__ B3) {
    int e = threadIdx.x;     // 96 threads
    float s = 0.f, ss = 0.f;
    for (int n = 0; n < Nn; n++) { float v = x[n * ED_ + e]; s += v; ss += v * v; }
    float m = s / (float)Nn;
    float var = ss / (float)Nn - m * m;
    float a = g[e] * rsqrtf(var + EPSB);
    A3[e] = a; B3[e] = b[e] - a * m;
}

// ------- 8) sender/receiver projections: sp = nf@Ws, rp = nf@Wr + fc_out_b -------
__global__ void k_proj(const float* __restrict__ x, const float* __restrict__ A3,
                       const float* __restrict__ B3, const float* __restrict__ fow,
                       const float* __restrict__ fob, float* __restrict__ sp,
                       float* __restrict__ rp) {
    __shared__ float nf[ED_];
    int n = blockIdx.x, e = threadIdx.x;   // 96 threads
    nf[e] = A3[e] * x[n * ED_ + e] + B3[e];
    __syncthreads();
    float as = 0.f, ar = 0.f;
    for (int f = 0; f < ED_; f++) {
        float v = nf[f];
        as = fmaf(v, fow[f * ED_ + e], as);
        ar = fmaf(v, fow[(ED_ + f) * ED_ + e], ar);
    }
    sp[n * ED_ + e] = as;
    rp[n * ED_ + e] = ar + fob[e];
}

// ------- 9) edges + hard gumbel-softmax -> adj [N,N], zero diagonal -------
__global__ void __launch_bounds__(256) k_edge(const float* __restrict__ sp,
                                              const float* __restrict__ rp,
                                              const float* __restrict__ fcc,
                                              const float* __restrict__ fccb,
                                              const float* __restrict__ uni,
                                              float* __restrict__ adj) {
    __shared__ float lr[16 * ED_], ls[16 * ED_], w0[ED_], w1[ED_];
    int i0 = blockIdx.y * 16, j0 = blockIdx.x * 16;
    int tid = threadIdx.x;
    for (int t = tid; t < 16 * ED_; t += 256) {
        int r = t / ED_, e = t - r * ED_;
        int i = i0 + r, j = j0 + r;
        lr[t] = (i < Nn) ? rp[i * ED_ + e] : 0.f;
        ls[t] = (j < Nn) ? sp[j * ED_ + e] : 0.f;
    }
    if (tid < ED_) { w0[tid] = fcc[2 * tid]; w1[tid] = fcc[2 * tid + 1]; }
    __syncthreads();
    int ti = tid / 16, tj = tid & 15;
    int i = i0 + ti, j = j0 + tj;
    if (i >= Nn || j >= Nn) return;
    float l0 = fccb[0], l1 = fccb[1];
    for (int e = 0; e < ED_; e++) {
        float v = fmaxf(lr[ti * ED_ + e] + ls[tj * ED_ + e], 0.f);
        l0 = fmaf(v, w0[e], l0);
        l1 = fmaf(v, w1[e], l1);
    }
    size_t eidx = (size_t)i * Nn + j;
    float u0 = uni[eidx * 2], u1 = uni[eidx * 2 + 1];
    float g0 = -logf(-logf(u0 + 1e-20f) + 1e-20f);
    float g1 = -logf(-logf(u1 + 1e-20f) + 1e-20f);
    float z0 = (l0 + g0) * 2.0f, z1 = (l1 + g1) * 2.0f;   // /TEMP, TEMP=0.5 (exact)
    float mz = fmaxf(z0, z1);
    float ex0 = expf(z0 - mz), ex1 = expf(z1 - mz);
    float y0 = ex0 / (ex0 + ex1);
    float h0 = (z0 >= z1) ? 1.f : 0.f;                    // argmax, first index on tie
    float y = (h0 - y0) + y0;                             // replicate ref arithmetic
    adj[eidx] = (i == j) ? 0.f : y;
}

// ---------------- launch ----------------
extern "C" void kernel_launch(void* const* d_in, const int* in_sizes, int n_in,
                              void* d_out, int out_size, void* d_ws, size_t ws_size,
                              hipStream_t stream) {
    (void)in_sizes; (void)n_in; (void)out_size; (void)ws_size;
    const float* node_feats = (const float*)d_in[1];
    const float* conv1_w    = (const float*)d_in[2];
    const float* conv1_b    = (const float*)d_in[3];
    const float* conv2_w    = (const float*)d_in[4];
    const float* conv2_b    = (const float*)d_in[5];
    const float* bn1_g      = (const float*)d_in[6];
    const float* bn1_b      = (const float*)d_in[7];
    const float* bn2_g      = (const float*)d_in[8];
    const float* bn2_b      = (const float*)d_in[9];
    const float* bn3_g      = (const float*)d_in[10];
    const float* bn3_b      = (const float*)d_in[11];
    const float* fc_w       = (const float*)d_in[12];
    const float* fc_b       = (const float*)d_in[13];
    const float* fc_out_w   = (const float*)d_in[14];
    const float* fc_out_b   = (const float*)d_in[15];
    const float* fc_cat_w   = (const float*)d_in[16];
    const float* fc_cat_b   = (const float*)d_in[17];
    const float* uniform    = (const float*)d_in[18];
    float* adj = (float*)d_out;

    // workspace layout (floats)
    float* W = (float*)d_ws;
    size_t off_y2 = 0;                                     // 307*16*5982
    size_t off_y1 = off_y2 + (size_t)Nn * C2_ * L2_;       // 307*8*5991
    size_t off_xT = off_y1 + (size_t)Nn * C1_ * L1_;       // 307*6000
    size_t off_sm = off_xT + (size_t)Nn * Tt;              // small arrays
    float* y2   = W + off_y2;
    float* y1   = W + off_y1;
    float* xT   = W + off_xT;
    float* a1   = W + off_sm;          // 8
    float* d1   = a1 + 8;              // 8
    float* a2   = d1 + 8;              // 16
    float* d2   = a2 + 16;             // 16
    float* wsum = d2 + 16;             // 16*96
    float* bfc  = wsum + C2_ * ED_;    // 96
    float* A3   = bfc + ED_;           // 96
    float* B3   = A3 + ED_;            // 96
    // y1 region is dead after conv2: reuse for GEMM partials and node features
    float* part = y1;                          // 16*320*96
    float* xnf  = part + (size_t)C2_ * MP_ * ED_;  // 307*96
    float* sp   = xnf + (size_t)Nn * ED_;          // 307*96
    float* rp   = sp + (size_t)Nn * ED_;           // 307*96

    // 1) transpose
    k_transpose<<<dim3((Tt + 31) / 32, (Nn + 31) / 32), dim3(32, 8), 0, stream>>>(node_feats, xT);
    // 2) conv1 + relu
    k_conv1<<<dim3((L1_ + 255) / 256, Nn), 256, 0, stream>>>(xT, conv1_w, conv1_b, y1);
    // 3) BN1 stats
    k_bnstats<<<dim3(C1_), 256, 0, stream>>>(y1, C1_, L1_, bn1_g, bn1_b, a1, d1);
    // 4) conv2(bn1) + relu
    k_conv2<<<dim3((L2_ + 255) / 256, Nn), 256, 0, stream>>>(y1, conv2_w, conv2_b, a1, d1, y2);
    // 5) BN2 stats
    k_bnstats<<<dim3(C2_), 256, 0, stream>>>(y2, C2_, L2_, bn2_g, bn2_b, a2, d2);
    // 6) fold BN2 shift into FC bias
    k_wsum<<<dim3(C2_), ED_, 0, stream>>>(fc_w, wsum);
    k_biasfc<<<1, ED_, 0, stream>>>(wsum, d2, fc_b, bfc);
    // 7) big GEMM via f32 WMMA (K split across 16 channels x 20 M-tiles)
    k_gemm_wmma<<<dim3(C2_, MP_ / 16), 192, 0, stream>>>(y2, fc_w, a2, part);
    // 8) reduce partials + bias + relu
    k_combine<<<(Nn * ED_ + 255) / 256, 256, 0, stream>>>(part, bfc, xnf);
    // 9) BN3
    k_bn3<<<1, ED_, 0, stream>>>(xnf, bn3_g, bn3_b, A3, B3);
    // 10) sender/receiver projections
    k_proj<<<Nn, ED_, 0, stream>>>(xnf, A3, B3, fc_out_w, fc_out_b, sp, rp);
    // 11) edges + hard gumbel softmax -> adjacency
    k_edge<<<dim3((Nn + 15) / 16, (Nn + 15) / 16), 256, 0, stream>>>(sp, rp, fc_cat_w,
                                                                     fc_cat_b, uniform, adj);
}